// GNN_ChebConv_6536940224661
// MI455X (gfx1250) — compile-verified
//
#include <hip/hip_runtime.h>
#include <math.h>

#define LNEPS 1e-5f

typedef float v2f __attribute__((ext_vector_type(2)));
typedef float v8f __attribute__((ext_vector_type(8)));

// ---------------- elementwise helpers ----------------
__global__ __launch_bounds__(256) void fill_kernel(float* p, float v, int n) {
    int i = blockIdx.x * blockDim.x + threadIdx.x;
    if (i < n) p[i] = v;
}

__global__ __launch_bounds__(256) void scalecopy_kernel(float* out, const float* __restrict__ in,
                                                        float a, int n) {
    int i = blockIdx.x * blockDim.x + threadIdx.x;
    if (i < n) out[i] = a * in[i];
}

// ---------------- graph norm ----------------
__global__ __launch_bounds__(256) void degree_kernel(const int* __restrict__ dst, float* deg, int E) {
    int e = blockIdx.x * blockDim.x + threadIdx.x;
    if (e < E) atomicAdd(&deg[dst[e]], 1.0f);
}

__global__ __launch_bounds__(256) void invsqrt_kernel(float* deg, int n) {
    int i = blockIdx.x * blockDim.x + threadIdx.x;
    if (i < n) {
        float d = deg[i];
        deg[i] = d > 0.0f ? rsqrtf(fmaxf(d, 1.0f)) : 0.0f;
    }
}

__global__ __launch_bounds__(256) void edgenorm_kernel(const int* __restrict__ src,
                                                       const int* __restrict__ dst,
                                                       const float* __restrict__ dis,
                                                       float* __restrict__ w, int E) {
    int e = blockIdx.x * blockDim.x + threadIdx.x;
    if (e < E) w[e] = -dis[src[e]] * dis[dst[e]];
}

// ---------------- sparse Laplacian matvec: out[dst] += scale*w_e*h[src] ----------------
__global__ __launch_bounds__(256) void prop_kernel(const float* __restrict__ h, float* out,
                                                   const int* __restrict__ src,
                                                   const int* __restrict__ dst,
                                                   const float* __restrict__ w,
                                                   float scale, int dimShift, int dim, int nWork) {
    int gid = blockIdx.x * blockDim.x + threadIdx.x;
    if (gid >= nWork) return;
    int e = gid >> dimShift;
    int c = (gid & ((1 << dimShift) - 1)) << 2;
    int s = src[e], d = dst[e];
    float we = scale * w[e];
    float4 v = *(const float4*)(h + (size_t)s * dim + c);
    float* op = out + (size_t)d * dim + c;
    atomicAdd(op + 0, we * v.x);
    atomicAdd(op + 1, we * v.y);
    atomicAdd(op + 2, we * v.z);
    atomicAdd(op + 3, we * v.w);
}

// ---------------- WMMA fp32 GEMM: out[N, NT*16] = A[N,kdim] @ B[kdim, NT*16] (+ epilogue) ----------------
// One wave per 16-row strip; NT 16x16 col tiles per wave (A fragment reused across NT WMMAs).
template <int NT>
__global__ __launch_bounds__(256) void gemm_wmma(const float* __restrict__ A,
                                                 const float* __restrict__ B,
                                                 int kdim, int nrows,
                                                 const float* accIn,          // nullable (+= previous)
                                                 const float* __restrict__ bias,   // nullable
                                                 const float* __restrict__ bn_g,   // nullable (BN implies ReLU)
                                                 const float* __restrict__ bn_b,
                                                 const float* __restrict__ bn_m,
                                                 const float* __restrict__ bn_v,
                                                 const float* __restrict__ residual, // nullable
                                                 float* out) {
    const int ncols = NT * 16;
    int wave = threadIdx.x >> 5;
    int lane = threadIdx.x & 31;
    int strip = blockIdx.x * 8 + wave;       // uniform per wave
    if (strip * 16 >= nrows) return;         // uniform exit -> EXEC all-1s for WMMA
    int m0 = strip * 16;
    int row = lane & 15;                     // A row / B,C,D column within tile
    int half = lane >> 4;                    // selects K pair (A/B) and M half (C/D)

    v8f acc[NT] = {};
    const float* Arow = A + (size_t)(m0 + row) * kdim + half * 2;
    for (int k = 0; k < kdim; k += 4) {
        v2f a;
        a.x = Arow[k];
        a.y = Arow[k + 1];
        int kb = k + half * 2;
        const float* B0 = B + (size_t)kb * ncols + row;
#pragma unroll
        for (int t = 0; t < NT; ++t) {
            v2f b;
            b.x = B0[t * 16];
            b.y = B0[t * 16 + ncols];
            acc[t] = __builtin_amdgcn_wmma_f32_16x16x4_f32(
                false, a, false, b, (short)0, acc[t], false, false);
        }
    }
#pragma unroll
    for (int t = 0; t < NT; ++t) {
        int col = t * 16 + row;
        float bsv = bias ? bias[col] : 0.0f;
        float sc = 1.0f, sh = 0.0f;
        bool dobn = (bn_g != nullptr);
        if (dobn) {
            float s = bn_g[col] * rsqrtf(bn_v[col] + LNEPS);
            sc = s;
            sh = bn_b[col] - bn_m[col] * s;
        }
#pragma unroll
        for (int j = 0; j < 8; ++j) {
            int m = m0 + j + half * 8;
            size_t idx = (size_t)m * ncols + col;
            float v = acc[t][j];
            if (accIn) v += accIn[idx];
            v += bsv;
            if (dobn) v = fmaxf(v * sc + sh, 0.0f);
            if (residual) v += residual[idx];
            out[idx] = v;
        }
    }
}

// ---------------- LayerNorm(+ReLU) over W features, one wave per row ----------------
template <int W>
__global__ __launch_bounds__(256) void ln_relu_kernel(float* h, const float* __restrict__ g,
                                                      const float* __restrict__ b, int nrows) {
    int wave = threadIdx.x >> 5, lane = threadIdx.x & 31;
    int row = blockIdx.x * 8 + wave;
    if (row >= nrows) return;
    float* hr = h + (size_t)row * W;
    float x0 = hr[lane];
    float x1 = (W == 64) ? hr[lane + 32] : 0.0f;
    float sum = x0 + x1;
    float sq = x0 * x0 + x1 * x1;
#pragma unroll
    for (int off = 16; off > 0; off >>= 1) {
        sum += __shfl_xor(sum, off, 32);
        sq += __shfl_xor(sq, off, 32);
    }
    float mu = sum / (float)W;
    float var = sq / (float)W - mu * mu;
    float rs = rsqrtf(var + LNEPS);
    hr[lane] = fmaxf((x0 - mu) * rs * g[lane] + b[lane], 0.0f);
    if (W == 64) hr[lane + 32] = fmaxf((x1 - mu) * rs * g[lane + 32] + b[lane + 32], 0.0f);
}

// ---------------- tiny heads ----------------
__global__ __launch_bounds__(256) void posw2_kernel(const float* __restrict__ p,
                                                    const float* __restrict__ w2,
                                                    const float* __restrict__ b2,
                                                    float* coords, int n) {
    int i = blockIdx.x * blockDim.x + threadIdx.x;
    if (i >= n) return;
    float c0 = b2[0], c1 = b2[1];
    const float* pr = p + (size_t)i * 64;
#pragma unroll
    for (int k = 0; k < 64; ++k) {
        float v = pr[k];
        c0 += v * w2[2 * k];
        c1 += v * w2[2 * k + 1];
    }
    coords[2 * i] = c0;
    coords[2 * i + 1] = c1;
}

__global__ __launch_bounds__(256) void radw2_kernel(const float* __restrict__ r,
                                                    const float* __restrict__ w2,
                                                    const float* __restrict__ b2,
                                                    float* rad, int n) {
    int i = blockIdx.x * blockDim.x + threadIdx.x;
    if (i >= n) return;
    float s = b2[0];
    const float* rr = r + (size_t)i * 32;
#pragma unroll
    for (int k = 0; k < 32; ++k) s += rr[k] * w2[k];
    rad[i] = 1.0f / (1.0f + expf(-s));
}

__global__ __launch_bounds__(256) void final_kernel(const float* __restrict__ coords,
                                                    const float* __restrict__ rad,
                                                    float* out, int n) {
    int i = blockIdx.x * blockDim.x + threadIdx.x;
    if (i >= n) return;
    float cx = coords[2 * i], cy = coords[2 * i + 1];
    float nrm = sqrtf(cx * cx + cy * cy) + 1e-8f;
    float q = rad[i] / nrm;
    out[2 * i] = cx * q;
    out[2 * i + 1] = cy * q;
}

// ---------------- host launch ----------------
extern "C" void kernel_launch(void* const* d_in, const int* in_sizes, int n_in,
                              void* d_out, int out_size, void* d_ws, size_t ws_size,
                              hipStream_t stream) {
    (void)n_in; (void)out_size; (void)ws_size;
    const float* x        = (const float*)d_in[0];
    const float* conv0_W  = (const float*)d_in[1];
    const float* conv0_b  = (const float*)d_in[2];
    const float* convs_W  = (const float*)d_in[3];
    const float* convs_b  = (const float*)d_in[4];
    const float* bn_g     = (const float*)d_in[5];
    const float* bn_b     = (const float*)d_in[6];
    const float* bn_m     = (const float*)d_in[7];
    const float* bn_v     = (const float*)d_in[8];
    const float* pos_w1   = (const float*)d_in[9];
    const float* pos_b1   = (const float*)d_in[10];
    const float* pos_ln_g = (const float*)d_in[11];
    const float* pos_ln_b = (const float*)d_in[12];
    const float* pos_w2   = (const float*)d_in[13];
    const float* pos_b2   = (const float*)d_in[14];
    const float* rad_w1   = (const float*)d_in[15];
    const float* rad_b1   = (const float*)d_in[16];
    const float* rad_ln_g = (const float*)d_in[17];
    const float* rad_ln_b = (const float*)d_in[18];
    const float* rad_w2   = (const float*)d_in[19];
    const float* rad_b2   = (const float*)d_in[20];
    const int*   eidx     = (const int*)d_in[21];

    const int N = in_sizes[0] / 128;
    const int E = in_sizes[21] / 2;
    const int* src = eidx;
    const int* dst = eidx + E;

    char* wsp = (char*)d_ws;
    auto take = [&](size_t bytes) -> float* {
        float* p = (float*)wsp;
        wsp += (bytes + 255) & ~(size_t)255;
        return p;
    };
    float* dis  = take((size_t)N * 4);          // degree -> d^{-1/2} in place
    float* wE   = take((size_t)E * 4);          // per-edge -d^{-1/2}_s d^{-1/2}_d
    float* T1   = take((size_t)N * 128 * 4);
    float* T2   = take((size_t)N * 128 * 4);
    float* accB = take((size_t)N * 64 * 4);
    float* hA   = take((size_t)N * 64 * 4);
    float* hB   = take((size_t)N * 64 * 4);
    // head buffers alias conv scratch (free after conv stack)
    float* pbuf   = accB;              // N x 64
    float* rbuf   = T1;                // N x 32
    float* coords = T2;                // N x 2
    float* rad    = T2 + (size_t)N * 2;

    dim3 blk(256);
    auto cdiv = [](int a, int b) { return (a + b - 1) / b; };

    // degree / normalization
    fill_kernel<<<cdiv(N, 256), blk, 0, stream>>>(dis, 0.0f, N);
    degree_kernel<<<cdiv(E, 256), blk, 0, stream>>>(dst, dis, E);
    invsqrt_kernel<<<cdiv(N, 256), blk, 0, stream>>>(dis, N);
    edgenorm_kernel<<<cdiv(E, 256), blk, 0, stream>>>(src, dst, dis, wE, E);

    const int strips = N / 16;          // N = 100000 = 6250*16
    const int gblocks = cdiv(strips, 8);

    // ---- Layer 0 (in dim 128) ----
    {
        gemm_wmma<4><<<gblocks, blk, 0, stream>>>(x, conv0_W, 128, N,
            nullptr, nullptr, nullptr, nullptr, nullptr, nullptr, nullptr, accB);
        fill_kernel<<<cdiv(N * 128, 256), blk, 0, stream>>>(T1, 0.0f, N * 128);
        int work = E * 32;
        prop_kernel<<<cdiv(work, 256), blk, 0, stream>>>(x, T1, src, dst, wE, 1.0f, 5, 128, work);
        gemm_wmma<4><<<gblocks, blk, 0, stream>>>(T1, conv0_W + 8192, 128, N,
            accB, nullptr, nullptr, nullptr, nullptr, nullptr, nullptr, accB);
        // T2 = 2*L*T1 - T0 : init with -T0, scatter with scale 2
        scalecopy_kernel<<<cdiv(N * 128, 256), blk, 0, stream>>>(T2, x, -1.0f, N * 128);
        prop_kernel<<<cdiv(work, 256), blk, 0, stream>>>(T1, T2, src, dst, wE, 2.0f, 5, 128, work);
        gemm_wmma<4><<<gblocks, blk, 0, stream>>>(T2, conv0_W + 16384, 128, N,
            accB, conv0_b, bn_g, bn_b, bn_m, bn_v, nullptr, hA);
    }

    // ---- Layers 1..2 (dim 64, residual) ----
    float* h = hA;
    float* hn = hB;
    for (int l = 1; l <= 2; ++l) {
        const float* W = convs_W + (size_t)(l - 1) * 3 * 64 * 64;
        const float* bias = convs_b + (size_t)(l - 1) * 64;
        gemm_wmma<4><<<gblocks, blk, 0, stream>>>(h, W, 64, N,
            nullptr, nullptr, nullptr, nullptr, nullptr, nullptr, nullptr, accB);
        int work = E * 16;
        fill_kernel<<<cdiv(N * 64, 256), blk, 0, stream>>>(T1, 0.0f, N * 64);
        prop_kernel<<<cdiv(work, 256), blk, 0, stream>>>(h, T1, src, dst, wE, 1.0f, 4, 64, work);
        gemm_wmma<4><<<gblocks, blk, 0, stream>>>(T1, W + 4096, 64, N,
            accB, nullptr, nullptr, nullptr, nullptr, nullptr, nullptr, accB);
        scalecopy_kernel<<<cdiv(N * 64, 256), blk, 0, stream>>>(T2, h, -1.0f, N * 64);
        prop_kernel<<<cdiv(work, 256), blk, 0, stream>>>(T1, T2, src, dst, wE, 2.0f, 4, 64, work);
        gemm_wmma<4><<<gblocks, blk, 0, stream>>>(T2, W + 8192, 64, N,
            accB, bias, bn_g + l * 64, bn_b + l * 64, bn_m + l * 64, bn_v + l * 64, h, hn);
        float* t = h; h = hn; hn = t;
    }

    // ---- position head ----
    gemm_wmma<4><<<gblocks, blk, 0, stream>>>(h, pos_w1, 64, N,
        nullptr, pos_b1, nullptr, nullptr, nullptr, nullptr, nullptr, pbuf);
    ln_relu_kernel<64><<<cdiv(N, 8), blk, 0, stream>>>(pbuf, pos_ln_g, pos_ln_b, N);
    posw2_kernel<<<cdiv(N, 256), blk, 0, stream>>>(pbuf, pos_w2, pos_b2, coords, N);

    // ---- radius head ----
    gemm_wmma<2><<<gblocks, blk, 0, stream>>>(h, rad_w1, 64, N,
        nullptr, rad_b1, nullptr, nullptr, nullptr, nullptr, nullptr, rbuf);
    ln_relu_kernel<32><<<cdiv(N, 8), blk, 0, stream>>>(rbuf, rad_ln_g, rad_ln_b, N);
    radw2_kernel<<<cdiv(N, 256), blk, 0, stream>>>(rbuf, rad_w2, rad_b2, rad, N);

    // ---- normalize_with_radius ----
    final_kernel<<<cdiv(N, 256), blk, 0, stream>>>(coords, rad, (float*)d_out, N);
}